// DensityCalculator_40596030882407
// MI455X (gfx1250) — compile-verified
//
#include <hip/hip_runtime.h>
#include <hip/hip_bf16.h>
#include <math.h>

// ---------------------------------------------------------------------------
// Problem constants (from reference)
// ---------------------------------------------------------------------------
#define N1D      64
#define GRID_N   (64 * 64 * 64)     // 262144
#define N_ATOMS  128
#define K_TERMS  6
#define SPACING  0.5f
#define LOG2E    1.4426950408889634f

typedef float v2f __attribute__((ext_vector_type(2)));
typedef float v8f __attribute__((ext_vector_type(8)));

// Raw hardware exp2: argument is always <= 0 here (bw < 0, d2 >= 0), so the
// OCML overflow fixup (cmp/cndmask/ldexp) is dead weight; v_exp_f32 flushes
// large-negative inputs to 0, which is exactly the Gaussian tail we want.
__device__ __forceinline__ float fast_exp2(float x) {
#if __has_builtin(__builtin_amdgcn_exp2f)
    return __builtin_amdgcn_exp2f(x);
#else
    return exp2f(x);
#endif
}

// ---------------------------------------------------------------------------
// Kernel 1: build shifted/scaled DFT matrices.
//  Wf[k][n] = (1/8)*(-1)^(k+n) * exp(-i*2*pi*k*n/64)   (fftshift folded in)
//  Wi = conj(Wf)  (inverse, ortho-normalized: 1/8 per axis both ways)
// ---------------------------------------------------------------------------
__global__ __launch_bounds__(256) void gen_dft_kernel(float* __restrict__ Wre,
                                                      float* __restrict__ WimF,
                                                      float* __restrict__ WimI) {
    int t = blockIdx.x * blockDim.x + threadIdx.x;
    if (t >= N1D * N1D) return;
    int k = t >> 6;
    int n = t & 63;
    int kn = (k * n) & 63;                       // exact mod-64 phase
    float ang = (float)kn * 0.09817477042468103f; // pi/32
    float sgn = ((k + n) & 1) ? -0.125f : 0.125f;
    float c = sgn * cosf(ang);
    float s = sgn * sinf(ang);
    Wre[t]  = c;
    WimF[t] = -s;   // forward: exp(-i.)
    WimI[t] = s;    // inverse: exp(+i.)
}

// ---------------------------------------------------------------------------
// Kernel 2: Gaussian density splat.
// One wave handles 16 consecutive grid points (z-fastest) x all 128 atoms.
// Cross term -2*g.x computed with V_WMMA_F32_16X16X4_F32 (M=grid, N=atom, K=xyz).
// Coordinates centered at box center (16,16,16) to reduce f32 cancellation.
// ---------------------------------------------------------------------------
__global__ __launch_bounds__(256) void density_kernel(const float* __restrict__ X,
                                                      const float* __restrict__ aw,
                                                      const float* __restrict__ bw,
                                                      float* __restrict__ rho_re,
                                                      float* __restrict__ rho_im) {
    const int lane   = threadIdx.x & 31;
    const int waveId = (blockIdx.x * blockDim.x + threadIdx.x) >> 5;
    const int g0 = waveId << 4;                 // 16-aligned base grid index
    const int xg = g0 >> 12;
    const int yg = (g0 >> 6) & 63;
    const int zb = g0 & 63;

    const float cx = 16.0f;                     // box center (64*0.5/2)
    const float gx = (float)xg * SPACING - cx;
    const float gy = (float)yg * SPACING - cx;
    const int half = lane >> 4;                 // 0: K={0,1}, 1: K={2,3}
    const int l16  = lane & 15;

    // A operand: A[m,k] = -2 * g_m[k]  (m = l16 across lanes, K split by half)
    float gzm = ((float)(zb + l16) * SPACING) - cx;
    v2f Aop;
    Aop.x = half ? (-2.0f * gzm) : (-2.0f * gx);
    Aop.y = half ? 0.0f          : (-2.0f * gy);

    // |g_m|^2 for this lane's 8 output rows: m = v + 8*half
    float gsq[8];
#pragma unroll
    for (int v = 0; v < 8; ++v) {
        float gz = ((float)(zb + v + 8 * half) * SPACING) - cx;
        gsq[v] = gx * gx + gy * gy + gz * gz;
    }

    float dens[8] = {0.f, 0.f, 0.f, 0.f, 0.f, 0.f, 0.f, 0.f};

    for (int a0 = 0; a0 < N_ATOMS; a0 += 16) {
        const int a = a0 + l16;                 // atom column owned by this lane
        float ax = X[3 * a + 0] - cx;
        float ay = X[3 * a + 1] - cx;
        float az = X[3 * a + 2] - cx;
        float xsq = ax * ax + ay * ay + az * az;

        // B operand: B[k,a] = x_a[k]  (rows K split across lane halves)
        v2f Bop;
        Bop.x = half ? az   : ax;
        Bop.y = half ? 0.0f : ay;

        v8f dot = {};
        dot = __builtin_amdgcn_wmma_f32_16x16x4_f32(
            false, Aop, false, Bop, (short)0, dot, false, false);

        float awk[K_TERMS], bl2[K_TERMS];
#pragma unroll
        for (int k = 0; k < K_TERMS; ++k) {
            awk[k] = aw[a * K_TERMS + k];
            bl2[k] = bw[a * K_TERMS + k] * LOG2E;   // use native exp2
        }

#pragma unroll
        for (int v = 0; v < 8; ++v) {
            float d2 = dot[v] + gsq[v] + xsq;
            float s = 0.0f;
#pragma unroll
            for (int k = 0; k < K_TERMS; ++k)
                s += awk[k] * fast_exp2(bl2[k] * d2);   // raw v_exp_f32
            dens[v] += s;
        }
    }

    // Sum over the 16 atom columns = sum across the 16 lanes of each half.
#pragma unroll
    for (int v = 0; v < 8; ++v) {
        float s = dens[v];
        s += __shfl_xor(s, 1, 32);
        s += __shfl_xor(s, 2, 32);
        s += __shfl_xor(s, 4, 32);
        s += __shfl_xor(s, 8, 32);
        dens[v] = s;
    }
    if (lane == 0) {
#pragma unroll
        for (int v = 0; v < 8; ++v) rho_re[g0 + v] = dens[v];
    }
    if (lane == 16) {
#pragma unroll
        for (int v = 0; v < 8; ++v) rho_re[g0 + 8 + v] = dens[v];
    }
    if (lane < 16) rho_im[g0 + lane] = 0.0f;    // imag part of rho = 0
}

// ---------------------------------------------------------------------------
// Kernel 3: one axis pass of the (shifted) DFT as a complex matmul.
//   U[k][p] = sum_n M[k,n] * T[p,n],  T: [4096][64], U: [64][4096]
// Storing U as [k][p] rotates the axes so 3 passes return to [x][y][z].
// 16x16 output tile per wave; K-loop of 16 x 4 f32 WMMAs.
// ---------------------------------------------------------------------------
__global__ __launch_bounds__(256) void dft_pass_kernel(const float* __restrict__ Tre,
                                                       const float* __restrict__ Tim,
                                                       const float* __restrict__ Mre,
                                                       const float* __restrict__ Mim,
                                                       float* __restrict__ Ure,
                                                       float* __restrict__ Uim) {
    const int lane   = threadIdx.x & 31;
    const int waveId = (blockIdx.x * blockDim.x + threadIdx.x) >> 5;  // 0..1023
    const int p0 = (waveId & 255) << 4;
    const int k0 = (waveId >> 8) << 4;
    const int half = lane >> 4;
    const int l16  = lane & 15;
    const int krow = k0 + l16;      // A-matrix row for this lane
    const int prow = p0 + l16;      // B column == T row for this lane

    v8f rr = {}, ii = {}, ri = {}, ir = {};

#pragma unroll 4
    for (int n0 = 0; n0 < N1D; n0 += 4) {
        const int koff = n0 + 2 * half;
        v2f mr = *(const v2f*)(Mre + krow * N1D + koff);
        v2f mi = *(const v2f*)(Mim + krow * N1D + koff);
        v2f tr = *(const v2f*)(Tre + prow * N1D + koff);
        v2f ti = *(const v2f*)(Tim + prow * N1D + koff);
        rr = __builtin_amdgcn_wmma_f32_16x16x4_f32(false, mr, false, tr, (short)0, rr, false, false);
        ii = __builtin_amdgcn_wmma_f32_16x16x4_f32(false, mi, false, ti, (short)0, ii, false, false);
        ri = __builtin_amdgcn_wmma_f32_16x16x4_f32(false, mr, false, ti, (short)0, ri, false, false);
        ir = __builtin_amdgcn_wmma_f32_16x16x4_f32(false, mi, false, tr, (short)0, ir, false, false);
    }

#pragma unroll
    for (int v = 0; v < 8; ++v) {
        const int row = k0 + v + 8 * half;
        const int idx = row * 4096 + p0 + l16;
        Ure[idx] = rr[v] - ii[v];
        Uim[idx] = ri[v] + ir[v];
    }
}

// ---------------------------------------------------------------------------
// Kernel 4: apply radial Hamming filter in Fourier space (in place).
// ---------------------------------------------------------------------------
__global__ __launch_bounds__(256) void hamming_kernel(float* __restrict__ re,
                                                      float* __restrict__ im,
                                                      const float* __restrict__ h) {
    int g = blockIdx.x * blockDim.x + threadIdx.x;
    if (g >= GRID_N) return;
    float hv = h[g];
    re[g] *= hv;
    im[g] *= hv;
}

// ---------------------------------------------------------------------------
// Kernels 5-7: deterministic normalize (zero mean, unit population std).
// ---------------------------------------------------------------------------
__global__ __launch_bounds__(256) void reduce1_kernel(const float* __restrict__ x,
                                                      float* __restrict__ psum,
                                                      float* __restrict__ psq) {
    __shared__ float ssum[256];
    __shared__ float ssq[256];
    const int tid = threadIdx.x;
    const int base = blockIdx.x * 512;
    float a = x[base + tid];
    float b = x[base + 256 + tid];
    ssum[tid] = a + b;
    ssq[tid]  = a * a + b * b;
    __syncthreads();
    for (int s = 128; s > 0; s >>= 1) {
        if (tid < s) { ssum[tid] += ssum[tid + s]; ssq[tid] += ssq[tid + s]; }
        __syncthreads();
    }
    if (tid == 0) { psum[blockIdx.x] = ssum[0]; psq[blockIdx.x] = ssq[0]; }
}

__global__ __launch_bounds__(256) void reduce2_kernel(const float* __restrict__ psum,
                                                      const float* __restrict__ psq,
                                                      float* __restrict__ stats) {
    __shared__ float ssum[256];
    __shared__ float ssq[256];
    const int tid = threadIdx.x;
    ssum[tid] = psum[tid] + psum[tid + 256];
    ssq[tid]  = psq[tid] + psq[tid + 256];
    __syncthreads();
    for (int s = 128; s > 0; s >>= 1) {
        if (tid < s) { ssum[tid] += ssum[tid + s]; ssq[tid] += ssq[tid + s]; }
        __syncthreads();
    }
    if (tid == 0) {
        const float invN = 1.0f / (float)GRID_N;
        float mean = ssum[0] * invN;
        float var  = fmaxf(ssq[0] * invN - mean * mean, 0.0f);
        stats[0] = mean;
        stats[1] = 1.0f / (sqrtf(var) + 1e-8f);
    }
}

__global__ __launch_bounds__(256) void finalize_kernel(const float* __restrict__ re,
                                                       const float* __restrict__ stats,
                                                       float* __restrict__ out) {
    int g = blockIdx.x * blockDim.x + threadIdx.x;
    if (g >= GRID_N) return;
    out[g] = (re[g] - stats[0]) * stats[1];
}

// ---------------------------------------------------------------------------
// Host orchestration (graph-capture safe: only kernel launches on `stream`).
// ---------------------------------------------------------------------------
extern "C" void kernel_launch(void* const* d_in, const int* in_sizes, int n_in,
                              void* d_out, int out_size, void* d_ws, size_t ws_size,
                              hipStream_t stream) {
    const float* X   = (const float*)d_in[0];   // (128,3)
    const float* aw  = (const float*)d_in[1];   // (128,6)
    const float* bw  = (const float*)d_in[2];   // (128,6)
    // d_in[3] real_grid_flat: recomputed analytically in-kernel (exact)
    const float* ham = (const float*)d_in[4];   // (262144,)
    float* out = (float*)d_out;

    float* ws   = (float*)d_ws;
    float* Wre  = ws;                    // 4096
    float* WimF = Wre + 4096;            // 4096
    float* WimI = WimF + 4096;           // 4096
    float* Are  = WimI + 4096;           // GRID_N
    float* Aim  = Are + GRID_N;
    float* Bre  = Aim + GRID_N;
    float* Bim  = Bre + GRID_N;
    float* psum = Bim + GRID_N;          // 512
    float* psq  = psum + 512;            // 512
    float* stats = psq + 512;            // 2

    // DFT matrices (shift + ortho scale folded in)
    gen_dft_kernel<<<16, 256, 0, stream>>>(Wre, WimF, WimI);

    // rho -> A (re), im = 0.   16384 waves, 16 grid pts each.
    density_kernel<<<2048, 256, 0, stream>>>(X, aw, bw, Are, Aim);

    // Forward 3D DFT (3 axis passes, axis-rotating complex WMMA matmuls)
    dft_pass_kernel<<<128, 256, 0, stream>>>(Are, Aim, Wre, WimF, Bre, Bim);
    dft_pass_kernel<<<128, 256, 0, stream>>>(Bre, Bim, Wre, WimF, Are, Aim);
    dft_pass_kernel<<<128, 256, 0, stream>>>(Are, Aim, Wre, WimF, Bre, Bim);

    // Radial Hamming low-pass (layout matches fftshifted frequency grid)
    hamming_kernel<<<GRID_N / 256, 256, 0, stream>>>(Bre, Bim, ham);

    // Inverse 3D DFT
    dft_pass_kernel<<<128, 256, 0, stream>>>(Bre, Bim, Wre, WimI, Are, Aim);
    dft_pass_kernel<<<128, 256, 0, stream>>>(Are, Aim, Wre, WimI, Bre, Bim);
    dft_pass_kernel<<<128, 256, 0, stream>>>(Bre, Bim, Wre, WimI, Are, Aim);

    // Normalize real part: (x - mean) / (std + 1e-8)
    reduce1_kernel<<<512, 256, 0, stream>>>(Are, psum, psq);
    reduce2_kernel<<<1, 256, 0, stream>>>(psum, psq, stats);
    finalize_kernel<<<GRID_N / 256, 256, 0, stream>>>(Are, stats, out);
}